// LengthRegulator_50989851738802
// MI455X (gfx1250) — compile-verified
//
#include <hip/hip_runtime.h>
#include <stdint.h>

// LengthRegulator for MI455X (gfx1250).
// B=32, T=512, D=256, MAX_DUR=4096. Pure data movement: 128 MB writes +
// ~16 MB reads (x lives in L2) @ 23.3 TB/s => ~6 us floor. The expansion is
// done with the CDNA5 async global->LDS->global copy path (ASYNCcnt),
// batched 8 transactions deep per wave with a single s_wait_asynccnt fence;
// trailing stores drain via the implicit S_WAIT_IDLE at S_ENDPGM.

#define B_   32
#define T_   512
#define D_   256
#define MAXD 4096

// ---------------- Kernel 1: inclusive scan of durations per batch ----------
__global__ void __launch_bounds__(512)
lr_scan_kernel(const int* __restrict__ dur, int* __restrict__ cum,
               int* __restrict__ total_out) {
    __shared__ int buf[2][T_];
    const int b = blockIdx.x;
    const int t = threadIdx.x;

    buf[0][t] = dur[b * T_ + t];
    __syncthreads();

    int src = 0;
    #pragma unroll
    for (int off = 1; off < T_; off <<= 1) {
        int v = buf[src][t];
        if (t >= off) v += buf[src][t - off];
        buf[src ^ 1][t] = v;
        src ^= 1;
        __syncthreads();
    }

    const int c = buf[src][t];
    cum[b * T_ + t] = c;
    if (t == T_ - 1) total_out[b] = c;   // int32 bit pattern into output tail
}

// ---------------- CDNA5 async copy helpers (inline asm, gfx1250) -----------
__device__ __forceinline__ void async_load_b128(unsigned lds_addr,
                                                const void* gaddr) {
    asm volatile("global_load_async_to_lds_b128 %0, %1, off"
                 :: "v"(lds_addr), "v"(gaddr) : "memory");
}
__device__ __forceinline__ void async_store_b128(void* gaddr,
                                                 unsigned lds_addr) {
    asm volatile("global_store_async_from_lds_b128 %0, %1, off"
                 :: "v"(gaddr), "v"(lds_addr) : "memory");
}
__device__ __forceinline__ void wait_async0() {
    asm volatile("s_wait_asynccnt 0" ::: "memory");
}

// ---------------- Kernel 2: expand rows by duration ------------------------
// grid = (MAXD/32, B), block = 256 threads (8 wave32).
// Each wave owns 4 frames and a 4 KB LDS staging area (4 x 1 KB rows).
// Phase 1: binary-search source rows, issue 8 async B128 loads (512 B each).
// Phase 2: one s_wait_asynccnt 0, then 8 async B128 stores.
// Invalid frames (frame >= total) are zero-filled with plain B128 stores.
__global__ void __launch_bounds__(256)
lr_expand_kernel(const float* __restrict__ x, const int* __restrict__ cum_g,
                 float* __restrict__ out) {
    __shared__ int cum[T_];
    __shared__ __align__(16) float wbuf[8][4][D_];   // 8 waves x 4 rows x 1 KB

    const int b    = blockIdx.y;
    const int tid  = threadIdx.x;
    const int wave = tid >> 5;
    const int lane = tid & 31;

    for (int i = tid; i < T_; i += 256) cum[i] = cum_g[b * T_ + i];
    __syncthreads();

    const int total = cum[T_ - 1];
    const int frame0 = blockIdx.x * 32 + wave * 4;   // wave-uniform

    // LDS byte offset of this wave's staging area (low 32 bits of the generic
    // address == workgroup-relative LDS offset on gfx1250).
    const unsigned lbase = (unsigned)(uintptr_t)(void*)&wbuf[wave][0][0]
                           + (unsigned)lane * 16u;

    // ---- Phase 1: gather source rows into LDS (8 async loads in flight) ----
    #pragma unroll
    for (int i = 0; i < 4; ++i) {
        const int frame = frame0 + i;
        if (frame < total) {
            // idx = min(searchsorted_right(cum, frame), T-1)
            int lo = 0, hi = T_;
            while (lo < hi) {
                const int mid = (lo + hi) >> 1;
                if (cum[mid] <= frame) lo = mid + 1; else hi = mid;
            }
            const int idx = lo < (T_ - 1) ? lo : (T_ - 1);

            const char* src = (const char*)(x + (((size_t)b * T_ + idx) * D_))
                              + lane * 16;
            const unsigned la = lbase + (unsigned)i * (D_ * 4u);
            async_load_b128(la,        src);
            async_load_b128(la + 512u, src + 512);
        } else {
            // dead frame: zero directly, no LDS round trip
            float4* dv = (float4*)(out + (((size_t)b * MAXD + frame) * D_));
            const float4 z = make_float4(0.f, 0.f, 0.f, 0.f);
            dv[lane]      = z;
            dv[lane + 32] = z;
        }
    }

    wait_async0();   // all 8 loads landed in LDS

    // ---- Phase 2: scatter rows to output (stores drain at S_ENDPGM) -------
    #pragma unroll
    for (int i = 0; i < 4; ++i) {
        const int frame = frame0 + i;
        if (frame < total) {
            char* dst = (char*)(out + (((size_t)b * MAXD + frame) * D_))
                        + lane * 16;
            const unsigned la = lbase + (unsigned)i * (D_ * 4u);
            async_store_b128(dst,       la);
            async_store_b128(dst + 512, la + 512u);
        }
    }
}

// ---------------- Launch ---------------------------------------------------
extern "C" void kernel_launch(void* const* d_in, const int* in_sizes, int n_in,
                              void* d_out, int out_size, void* d_ws,
                              size_t ws_size, hipStream_t stream) {
    const float* x   = (const float*)d_in[0];
    const int*   dur = (const int*)d_in[1];
    // d_in[2] = max_dur scalar (4096), baked in as MAXD.

    float* out       = (float*)d_out;
    int*   total_out = (int*)d_out + (size_t)B_ * MAXD * D_;  // tuple tail
    int*   cum       = (int*)d_ws;                            // 64 KB scratch

    lr_scan_kernel<<<dim3(B_), dim3(T_), 0, stream>>>(dur, cum, total_out);
    lr_expand_kernel<<<dim3(MAXD / 32, B_), dim3(256), 0, stream>>>(x, cum, out);
}